// MultiHeadDecoder_20899310862618
// MI455X (gfx1250) — compile-verified
//
#include <hip/hip_runtime.h>
#include <hip/hip_bf16.h>
#include <math.h>

// MI455X (gfx1250) implementation.
// Roofline: dominant cost is one streaming pass over q (50000x512 f32 = 102.4MB)
// -> ~4.4us at 23.3 TB/s. Algebraic rewrite compat[n] = q[n] . (W_key @ Q)
// turns the 26-GFLOP GEMM into a memory-bound GEMV. WMMA (f32 16x16x4) is used
// for the two small 512x512 GEMVs in the prep phase; the hot loop uses wide
// non-temporal b128 loads + wave32 shuffle reductions.

typedef float __attribute__((ext_vector_type(2))) v2f;
typedef float __attribute__((ext_vector_type(4))) v4f;
typedef float __attribute__((ext_vector_type(8))) v8f;

#define NROWS 50000
#define DIM 512
#define NORM_FACTOR 0.04419417382415922f   // 1/sqrt(512)

// ---- float <-> order-preserving uint (for atomicMax argmax) ----
__device__ __forceinline__ unsigned int f2ord(float f) {
    unsigned int u = __float_as_uint(f);
    return (u & 0x80000000u) ? ~u : (u | 0x80000000u);
}

// ============================================================================
// Kernel 1: prep. One block, 1024 threads (32 waves).
//   concat    = Wc0*l + Wc1*context          -> d_out[514..1025]
//   g_context = Wg0*g + Wg1*concat           -> LDS
//   Q = g_context @ W_query  (WMMA f32 16x16x4, wave w owns Q[16w..16w+15])
//   t = W_key @ Q            (WMMA f32 16x16x4, wave w owns t[16w..16w+15])
// A-matrix layout per ISA 7.12.2 (32-bit A 16x4): lanes 0-15 hold K=0,1 in
// VGPR0,1 for M=lane; lanes 16-31 hold K=2,3 for M=lane-16. B columns are all
// identical (vector broadcast), so every D column holds the GEMV result.
// ============================================================================
__global__ void prep_kernel(const float* __restrict__ l, const float* __restrict__ ctx,
                            const float* __restrict__ g,
                            const float* __restrict__ Wc, const float* __restrict__ Wg,
                            const float* __restrict__ Wq, const float* __restrict__ Wk,
                            float* __restrict__ out, float* __restrict__ ws_t,
                            float* __restrict__ ws_sum, unsigned long long* __restrict__ ws_pack)
{
    __shared__ float s_g[DIM];   // g_context
    __shared__ float s_Q[DIM];   // Q
    const int tid = threadIdx.x;

    if (tid < DIM) {
        float c0 = Wc[0] * l[tid] + Wc[1] * ctx[tid];
        out[514 + tid] = c0;                               // concat output
        s_g[tid] = Wg[0] * g[tid] + Wg[1] * c0;            // g_context
    }
    if (tid == 0) { *ws_sum = 0.0f; *ws_pack = 0ull; }     // reset reductions
    __syncthreads();

    const int wave = tid >> 5;
    const int lane = tid & 31;
    const int m    = lane & 15;              // row within 16-tile
    const int k0   = (lane >> 4) << 1;       // 0 for lanes 0-15, 2 for lanes 16-31
    const int jb   = wave * 16;

    // ---- Q[j] = sum_i g_context[i] * Wq[i*512 + j] ----
    v8f acc = {0.f,0.f,0.f,0.f,0.f,0.f,0.f,0.f};
    for (int kb = 0; kb < DIM; kb += 4) {
        v2f a, b;
        a.x = Wq[(size_t)(kb + k0)     * DIM + jb + m];    // A[m][k0]
        a.y = Wq[(size_t)(kb + k0 + 1) * DIM + jb + m];    // A[m][k0+1]
        b.x = s_g[kb + k0];                                // B broadcast column
        b.y = s_g[kb + k0 + 1];
        acc = __builtin_amdgcn_wmma_f32_16x16x4_f32(false, a, false, b,
                                                    (short)0, acc, false, false);
    }
    // C/D layout: VGPR r: lanes 0-15 -> M=r, lanes 16-31 -> M=8+r (N = lane%16)
    if (lane == 0) {
        #pragma unroll
        for (int r = 0; r < 8; ++r) s_Q[jb + r] = acc[r];
    }
    if (lane == 16) {
        #pragma unroll
        for (int r = 0; r < 8; ++r) s_Q[jb + 8 + r] = acc[r];
    }
    __syncthreads();

    // ---- t[i] = sum_j Wk[i*512 + j] * Q[j]  (A rows contiguous -> v2f loads) ----
    v8f acc2 = {0.f,0.f,0.f,0.f,0.f,0.f,0.f,0.f};
    const int ib = wave * 16;
    for (int kb = 0; kb < DIM; kb += 4) {
        v2f a = *(const v2f*)(Wk + (size_t)(ib + m) * DIM + kb + k0);
        v2f b;
        b.x = s_Q[kb + k0];
        b.y = s_Q[kb + k0 + 1];
        acc2 = __builtin_amdgcn_wmma_f32_16x16x4_f32(false, a, false, b,
                                                     (short)0, acc2, false, false);
    }
    if (lane == 0) {
        #pragma unroll
        for (int r = 0; r < 8; ++r) ws_t[ib + r] = acc2[r];
    }
    if (lane == 16) {
        #pragma unroll
        for (int r = 0; r < 8; ++r) ws_t[ib + 8 + r] = acc2[r];
    }
}

// ============================================================================
// Kernel 2: streaming GEMV + fused softmax-sum & argmax reductions.
// One wave32 per row: each lane does 4x float4 NT loads (row = 32 lanes *
// 16 floats, fully coalesced 512B segments), dot with t (staged in LDS),
// shfl_xor wave reduction. compat = tanh(s/sqrt(D))*10 <= 10, so
// exp(compat-10) is stable without a max pass. Argmax packs (ordered_value,
// ~row) into a u64 so atomicMax gives max value with first-index tie-break.
// ============================================================================
__global__ void gemv_reduce_kernel(const float* __restrict__ q, const int* __restrict__ mask,
                                   const float* __restrict__ ws_t,
                                   float* __restrict__ ws_sum,
                                   unsigned long long* __restrict__ ws_pack)
{
    __shared__ float s_t[DIM];
    __shared__ float s_e[8];
    __shared__ unsigned long long s_p[8];

    const int tid = threadIdx.x;
    s_t[tid]       = ws_t[tid];
    s_t[tid + 256] = ws_t[tid + 256];
    __syncthreads();

    const int lane = tid & 31;
    const int wave = tid >> 5;
    const int row  = blockIdx.x * 8 + wave;

    float e = 0.0f;
    unsigned long long pack = 0ull;

    if (row < NROWS) {
        const float* qr = q + (size_t)row * DIM;
        float acc = 0.0f;
        #pragma unroll
        for (int i = 0; i < 4; ++i) {
            const int off = i * 128 + lane * 4;
            v4f qv = __builtin_nontemporal_load((const v4f*)(qr + off)); // read-once stream
            acc += qv.x * s_t[off] + qv.y * s_t[off + 1]
                 + qv.z * s_t[off + 2] + qv.w * s_t[off + 3];
        }
        #pragma unroll
        for (int o = 16; o > 0; o >>= 1) acc += __shfl_xor(acc, o, 32);

        if (lane == 0 && mask[row] > 0) {
            float compat = tanhf(acc * NORM_FACTOR) * 10.0f;
            e = expf(compat - 10.0f);
            pack = ((unsigned long long)f2ord(compat) << 32)
                 | (unsigned long long)(0xFFFFFFFFu - (unsigned int)row);
        }
    }
    if (lane == 0) { s_e[wave] = e; s_p[wave] = pack; }
    __syncthreads();

    if (tid == 0) {
        float esum = 0.0f;
        unsigned long long pm = 0ull;
        #pragma unroll
        for (int w = 0; w < 8; ++w) { esum += s_e[w]; if (s_p[w] > pm) pm = s_p[w]; }
        if (esum != 0.0f) atomicAdd(ws_sum, esum);
        if (pm != 0ull)   atomicMax(ws_pack, pm);
    }
}

// ============================================================================
// Kernel 3: finalize. Grid covers 50000 for mask copy; block 0 gathers
// q[idx], recomputes compat[idx] (handles both argmax and is_random paths),
// writes attn/log_attn/idx.
// Output layout (flat, floats): q_max[0..511], attn[512], log_attn[513],
// concat[514..1025], mask[1026..51025], max_indx[51026].
// ============================================================================
__global__ void finalize_kernel(const float* __restrict__ q, const int* __restrict__ mask,
                                const int* __restrict__ is_random,
                                const long long* __restrict__ random_net,
                                const float* __restrict__ ws_t, const float* __restrict__ ws_sum,
                                const unsigned long long* __restrict__ ws_pack,
                                float* __restrict__ out)
{
    const int tid = threadIdx.x;
    const int gid = blockIdx.x * blockDim.x + tid;
    if (gid < NROWS) out[1026 + gid] = (float)mask[gid];   // mask_copy

    if (blockIdx.x == 0) {                                  // blockDim == 512
        __shared__ float s_red[DIM];
        const unsigned long long pack = *ws_pack;
        const int idx = (is_random[0] != 0)
                      ? (int)random_net[0]
                      : (int)(0xFFFFFFFFu - (unsigned int)(pack & 0xFFFFFFFFull));

        const float qv = q[(size_t)idx * DIM + tid];
        out[tid] = qv;                                      // q_max
        s_red[tid] = qv * ws_t[tid];
        __syncthreads();
        #pragma unroll
        for (int s = 256; s > 0; s >>= 1) {
            if (tid < s) s_red[tid] += s_red[tid + s];
            __syncthreads();
        }
        if (tid == 0) {
            const float ssum = *ws_sum;
            const float compat = (mask[idx] > 0)
                               ? tanhf(s_red[0] * NORM_FACTOR) * 10.0f
                               : -INFINITY;
            const float e = expf(compat - 10.0f);
            out[512]   = e / ssum;                          // attn_max
            out[513]   = (compat - 10.0f) - logf(ssum);     // log_attn_max
            out[51026] = (float)idx;                        // max_indx
        }
    }
}

extern "C" void kernel_launch(void* const* d_in, const int* in_sizes, int n_in,
                              void* d_out, int out_size, void* d_ws, size_t ws_size,
                              hipStream_t stream)
{
    const float*     q          = (const float*)d_in[0];
    const float*     l          = (const float*)d_in[1];
    const float*     ctx        = (const float*)d_in[2];
    const float*     g          = (const float*)d_in[3];
    const int*       mask       = (const int*)d_in[4];
    const int*       is_random  = (const int*)d_in[5];
    const long long* random_net = (const long long*)d_in[6];
    const float*     Wc         = (const float*)d_in[7];
    const float*     Wg         = (const float*)d_in[8];
    const float*     Wq         = (const float*)d_in[9];
    const float*     Wk         = (const float*)d_in[10];
    float* out = (float*)d_out;

    // Workspace: t[512] floats (2048B), sum float @2048, pack u64 @2056.
    float* ws_t   = (float*)d_ws;
    float* ws_sum = (float*)((char*)d_ws + 2048);
    unsigned long long* ws_pack = (unsigned long long*)((char*)d_ws + 2056);

    prep_kernel<<<1, 1024, 0, stream>>>(l, ctx, g, Wc, Wg, Wq, Wk,
                                        out, ws_t, ws_sum, ws_pack);
    gemv_reduce_kernel<<<(NROWS + 7) / 8, 256, 0, stream>>>(q, mask, ws_t, ws_sum, ws_pack);
    finalize_kernel<<<(NROWS + 511) / 512, 512, 0, stream>>>(q, mask, is_random, random_net,
                                                             ws_t, ws_sum, ws_pack, out);
}